// GraphConv_26860725469245
// MI455X (gfx1250) — compile-verified
//
#include <hip/hip_runtime.h>

// GraphConv for MI455X (gfx1250, wave32).
//   out = verts@W0^T + b0 + scatter_add(verts@W1^T + b1 over undirected edges)
// Kernel 1: dual fp32 WMMA GEMM (v_wmma_f32_16x16x4_f32), weights staged in LDS.
// Kernel 2: per-edge gather + hardware f32 atomic scatter (L2-resident tables).

typedef float v2f __attribute__((ext_vector_type(2)));
typedef float v4f __attribute__((ext_vector_type(4)));
typedef float v8f __attribute__((ext_vector_type(8)));

#define FEAT 64

// ---------------------------------------------------------------------------
// Kernel 1: y0 = x @ W0^T + b0  -> outW0 (d_out)
//           y1 = x @ W1^T + b1  -> outW1 (d_ws)
// One wave per 16-row tile; 8 waves (128 rows) per 256-thread block.
// K = 64 -> 16 WMMA k-steps of K=4; N = 64 -> 4 n-tiles of 16.
// ---------------------------------------------------------------------------
__global__ __launch_bounds__(256)
void gemm_dual_wmma(const float* __restrict__ verts,
                    const float* __restrict__ w0,
                    const float* __restrict__ b0,
                    const float* __restrict__ w1,
                    const float* __restrict__ b1,
                    float* __restrict__ outW0,
                    float* __restrict__ outW1,
                    int nV)
{
    __shared__ float lw0[FEAT * FEAT];   // 16 KB
    __shared__ float lw1[FEAT * FEAT];   // 16 KB  (well under 320 KB/WGP)

    // Cooperative stage of both weight matrices into LDS (float4 copies).
    for (int i = threadIdx.x; i < (FEAT * FEAT) / 4; i += 256) {
        ((v4f*)lw0)[i] = ((const v4f*)w0)[i];
        ((v4f*)lw1)[i] = ((const v4f*)w1)[i];
    }
    __syncthreads();   // before any wave can exit -> barrier always completes

    const int lane    = threadIdx.x & 31;
    const int wave    = threadIdx.x >> 5;
    const int nTiles  = (nV + 15) >> 4;
    const int rowTile = blockIdx.x * 8 + wave;     // uniform per wave
    if (rowTile >= nTiles) return;                 // whole-wave exit: EXEC stays all-1s
    const int row0 = rowTile * 16;

    // A-fragment addressing (16x4 f32, 2 VGPRs):
    //   lanes 0-15  hold M=lane,    K = kbase+0, kbase+1
    //   lanes 16-31 hold M=lane-16, K = kbase+2, kbase+3
    const int m  = lane & 15;
    const int kg = lane >> 4;                      // 0 or 1
    const int rA = min(row0 + m, nV - 1);          // clamp for (unused here) ragged tail
    const float* arow = verts + (size_t)rA * FEAT + kg * 2;

    v8f acc0[4] = {};
    v8f acc1[4] = {};

#pragma unroll
    for (int kk = 0; kk < 16; ++kk) {
        v2f a = *(const v2f*)(arow + kk * 4);
#pragma unroll
        for (int nt = 0; nt < 4; ++nt) {
            // B[k][n] = W[n][k]  (y = x @ W^T), so a lane reads 2 contiguous
            // floats of row n of W: single ds_load_b64 per fragment.
            const int woff = (nt * 16 + m) * FEAT + kk * 4 + kg * 2;
            v2f bb0 = *(const v2f*)(lw0 + woff);
            v2f bb1 = *(const v2f*)(lw1 + woff);
            acc0[nt] = __builtin_amdgcn_wmma_f32_16x16x4_f32(
                false, a, false, bb0, (short)0, acc0[nt], false, false);
            acc1[nt] = __builtin_amdgcn_wmma_f32_16x16x4_f32(
                false, a, false, bb1, (short)0, acc1[nt], false, false);
        }
    }

    // C/D layout: VGPR j holds M = j (lanes 0-15) or M = 8+j (lanes 16-31).
    const int mbase = row0 + kg * 8;
#pragma unroll
    for (int nt = 0; nt < 4; ++nt) {
        const int col   = nt * 16 + m;
        const float bi0 = b0[col];
        const float bi1 = b1[col];
#pragma unroll
        for (int j = 0; j < 8; ++j) {
            const int r = mbase + j;
            if (r < nV) {
                outW0[(size_t)r * FEAT + col] = acc0[nt][j] + bi0;
                outW1[(size_t)r * FEAT + col] = acc1[nt][j] + bi1;
            }
        }
    }
}

// ---------------------------------------------------------------------------
// Kernel 2: undirected scatter-add.  One wave per edge; lane owns 2 floats.
//   out[u] += h[v];  out[v] += h[u];   (hardware global_atomic_add_f32)
// h (25.6 MB) and out (25.6 MB) both fit in the 192 MB L2 -> L2-resident.
// ---------------------------------------------------------------------------
__global__ __launch_bounds__(256)
void edge_scatter(const int* __restrict__ edges,
                  const float* __restrict__ h,
                  float* __restrict__ out,
                  int nE)
{
    const int lane = threadIdx.x & 31;
    const int wave = threadIdx.x >> 5;
    const int e    = blockIdx.x * 8 + wave;        // uniform per wave
    if (e >= nE) return;

    const int u = edges[2 * e];
    const int v = edges[2 * e + 1];
    const int c = lane * 2;                        // 32 lanes x 2 floats = 64 cols

    v2f hu = *(const v2f*)(h + (size_t)u * FEAT + c);
    v2f hv = *(const v2f*)(h + (size_t)v * FEAT + c);

    float* pu = out + (size_t)u * FEAT + c;
    float* pv = out + (size_t)v * FEAT + c;

    unsafeAtomicAdd(pu,     hv.x);
    unsafeAtomicAdd(pu + 1, hv.y);
    unsafeAtomicAdd(pv,     hu.x);
    unsafeAtomicAdd(pv + 1, hu.y);
}

// ---------------------------------------------------------------------------
extern "C" void kernel_launch(void* const* d_in, const int* in_sizes, int n_in,
                              void* d_out, int out_size, void* d_ws, size_t ws_size,
                              hipStream_t stream)
{
    const float* verts = (const float*)d_in[0];   // (V, 64) f32
    const int*   edges = (const int*)  d_in[1];   // (E, 2)  int
    const float* w0w   = (const float*)d_in[2];   // (64, 64) f32
    const float* w0b   = (const float*)d_in[3];   // (64,)    f32
    const float* w1w   = (const float*)d_in[4];   // (64, 64) f32
    const float* w1b   = (const float*)d_in[5];   // (64,)    f32

    float* out = (float*)d_out;                   // (V, 64) f32
    float* h   = (float*)d_ws;                    // scratch: V*64 f32 = 25.6 MB

    const int nV = in_sizes[0] / FEAT;
    const int nE = in_sizes[1] / 2;

    const int nTiles = (nV + 15) >> 4;            // 16-row tiles
    dim3 blk(256);
    dim3 g1((nTiles + 7) / 8);                    // 8 waves (tiles) per block
    gemm_dual_wmma<<<g1, blk, 0, stream>>>(verts, w0w, w0b, w1w, w1b, out, h, nV);

    dim3 g2((nE + 7) / 8);                        // 8 edges (waves) per block
    edge_scatter<<<g2, blk, 0, stream>>>(edges, h, out, nE);
}